// memristor_dense_9113920602694
// MI455X (gfx1250) — compile-verified
//
#include <hip/hip_runtime.h>

// ---- problem constants (from reference) ----
static constexpr int B_     = 64;
static constexpr int NIN    = 1024;
static constexpr int NOUT   = 1024;
static constexpr int KROWS  = NIN + 1;   // 1025 (bias row folded)
static constexpr int KP     = 1028;      // padded to multiple of 4
static constexpr int KT     = KP / 4;    // 257 k-tiles of 4
static constexpr int NTERM  = 4;         // Taylor terms in 2^(delta*L)
static constexpr int KSPLIT = 16;        // K-chunks for wave-level parallelism
static constexpr int KCHUNK = (KT + KSPLIT - 1) / KSPLIT;  // 17 k-tiles per chunk

#define G_MIN_C 0.1f
#define G_SPAN  0.9f              /* G_MAX - G_MIN */
#define EBAR    1.1243281f        /* log2(2.18) */
#define LN2_C   0.69314718f

typedef float v2f __attribute__((ext_vector_type(2)));
typedef float v8f __attribute__((ext_vector_type(8)));

// workspace layout (bytes):
//   [0,16)                      : max|W| bits
//   [16, 16+1052672)            : A fragments (NTERM*KT*4*32*2 floats)
//   [1052688, +4MB)             : KSPLIT partial outputs (KSPLIT*64*1024 floats)
static constexpr size_t AF_OFF   = 16;
static constexpr size_t AF_FLTS  = (size_t)NTERM * KT * 4 * 32 * 2;     // 263168
static constexpr size_t PART_OFF = AF_OFF + AF_FLTS * 4;                 // 1052688
static constexpr int    PART_STRIDE = B_ * NOUT;                          // 65536

// ---------------- kernel 1: init workspace scalar ----------------
__global__ void init_ws_k(unsigned* ws) {
    // baseline max|W| includes bias entries (0.5)
    ws[0] = __float_as_uint(0.5f);
}

// ---------------- kernel 2: max|W| reduction ----------------
__global__ __launch_bounds__(256)
void maxabs_k(const float* __restrict__ wpos, const float* __restrict__ wneg,
              unsigned* __restrict__ ws) {
    const int n = NIN * NOUT;
    float m = 0.0f;
    for (int i = blockIdx.x * blockDim.x + threadIdx.x; i < n;
         i += gridDim.x * blockDim.x) {
        m = fmaxf(m, fmaxf(fabsf(wpos[i]), fabsf(wneg[i])));
    }
    // wave32 butterfly reduction, then one atomic per wave
    for (int off = 16; off > 0; off >>= 1)
        m = fmaxf(m, __shfl_xor(m, off, 32));
    if ((threadIdx.x & 31) == 0)
        atomicMax(ws, __float_as_uint(m));   // positive floats: bit order == value order
}

// ---------------- kernel 3: build A fragments ----------------
// Layout matches V_WMMA_F32_16X16X4_F32 A (16x4) lane layout exactly:
// af[((t*KT + ktile)*4 + mtile)*64 + lane*2 + j]
//   = A_t[ m = mtile*16 + (lane&15) ][ k = ktile*4 + (lane>>4)*2 + j ]
__global__ __launch_bounds__(256)
void build_afrag_k(const float* __restrict__ x, const unsigned* __restrict__ ws,
                   float* __restrict__ af) {
    const int total = (int)AF_FLTS;
    int idx = blockIdx.x * blockDim.x + threadIdx.x;
    if (idx >= total) return;
    int j     = idx & 1;
    int lane  = (idx >> 1) & 31;
    int mtile = (idx >> 6) & 3;
    int rest  = idx >> 8;
    int ktile = rest % KT;
    int t     = rest / KT;
    int b = mtile * 16 + (lane & 15);
    int k = ktile * 4 + ((lane >> 4) << 1) + j;

    float val = 0.0f;
    if (k < KROWS) {
        float xv = (k < NIN) ? x[b * NIN + k] : 1.0f;   // ones column (bias row)
        float ax = fabsf(xv);
        if (ax > 1e-30f) {
            float maxw  = __uint_as_float(ws[0]);
            float scale = 0.5f * maxw / G_SPAN;         // V_ref/(k_V*k_G)
            float L = __log2f(2.0f * ax);               // log2(|V|/V_ref)
            float s = (xv > 0.0f) ? 1.0f : -1.0f;
            float P = s * __builtin_exp2f(EBAR * L) * scale;
            float u = L * LN2_C;
            float pw = 1.0f;
            if      (t == 1) pw = u;
            else if (t == 2) pw = 0.5f * u * u;
            else if (t == 3) pw = (1.0f / 6.0f) * u * u * u;
            val = P * pw;
        }
    }
    af[idx] = val;
}

// ---------------- kernel 4: fused B-build + WMMA GEMM ----------------
// 1024 waves: 64 column-stripes x 16 K-chunks. Each wave computes a
// 64-row x 16-col partial tile over its K-chunk and writes it to a
// private partial-sum slab (no atomics -> deterministic).
__global__ __launch_bounds__(128)
void mem_gemm_k(const float* __restrict__ wpos, const float* __restrict__ wneg,
                const float* __restrict__ bpos, const float* __restrict__ bneg,
                const float* __restrict__ npar, const float* __restrict__ af,
                const unsigned* __restrict__ ws, float* __restrict__ part) {
    const int lane   = threadIdx.x & 31;
    const int chunk  = blockIdx.x & (KSPLIT - 1);                 // waves in a block share chunk
    const int stripe = (blockIdx.x >> 4) * 4 + (threadIdx.x >> 5);
    const int hf     = lane >> 4;                                 // which half of the wave
    const int n      = stripe * 16 + (lane & 15);
    const float kG   = G_SPAN / __uint_as_float(ws[0]);

    const int kt_beg = chunk * KCHUNK;
    const int kt_end = (kt_beg + KCHUNK < KT) ? kt_beg + KCHUNK : KT;

    v8f acc[4] = {};   // 4 m-tiles of 16x16 fp32 accumulators

    for (int ktile = kt_beg; ktile < kt_end; ++ktile) {
        // Each lane supplies B[k][n] for two k's: k = ktile*4 + hf*2 + {0,1}
        float c0[2], c1[2], c2[2], c3[2];
#pragma unroll
        for (int jj = 0; jj < 2; ++jj) {
            int k = ktile * 4 + hf * 2 + jj;
            float t0 = 0.f, t1 = 0.f, t2 = 0.f, t3 = 0.f;
            if (k < KROWS) {
                float wp, wn;
                if (k < NIN) { wp = wpos[k * NOUT + n]; wn = wneg[k * NOUT + n]; }
                else         { wp = bpos[n];            wn = bneg[n]; }
                v2f np = *(const v2f*)(npar + (size_t)k * (2 * NOUT) + 2 * n);
                float Gp = G_MIN_C + kG * wp;
                float Gn = G_MIN_C + kG * wn;
                float dp = __log2f(np[0]) - EBAR;
                float dn = __log2f(np[1]) - EBAR;
                float dp2 = dp * dp, dn2 = dn * dn;
                t0 = Gp - Gn;                       // differential folded into B
                t1 = Gp * dp  - Gn * dn;
                t2 = Gp * dp2 - Gn * dn2;
                t3 = Gp * dp2 * dp - Gn * dn2 * dn;
            }
            c0[jj] = t0; c1[jj] = t1; c2[jj] = t2; c3[jj] = t3;
        }
        v2f Bf[NTERM];
        Bf[0][0] = c0[0]; Bf[0][1] = c0[1];
        Bf[1][0] = c1[0]; Bf[1][1] = c1[1];
        Bf[2][0] = c2[0]; Bf[2][1] = c2[1];
        Bf[3][0] = c3[0]; Bf[3][1] = c3[1];

#pragma unroll
        for (int t = 0; t < NTERM; ++t) {
            const float* abase = af + (size_t)(t * KT + ktile) * 256 + lane * 2;
#pragma unroll
            for (int mt = 0; mt < 4; ++mt) {
                v2f a = *(const v2f*)(abase + mt * 64);
                acc[mt] = __builtin_amdgcn_wmma_f32_16x16x4_f32(
                    false, a, false, Bf[t], (short)0, acc[mt], false, false);
            }
        }
    }

    // Epilogue: C/D layout — VGPR j holds row (hf*8 + j), col = lane&15.
    // Unique (chunk,row,col) slot per lane-element: no atomics needed.
    float* pbase = part + (size_t)chunk * PART_STRIDE;
#pragma unroll
    for (int mt = 0; mt < 4; ++mt) {
#pragma unroll
        for (int j = 0; j < 8; ++j) {
            int row = mt * 16 + hf * 8 + j;
            pbase[(size_t)row * NOUT + n] = acc[mt][j];
        }
    }
}

// ---------------- kernel 5: deterministic partial reduction ----------------
__global__ __launch_bounds__(256)
void reduce_out_k(const float* __restrict__ part, float* __restrict__ out) {
    int idx = blockIdx.x * blockDim.x + threadIdx.x;
    if (idx >= PART_STRIDE) return;
    float s = 0.0f;
#pragma unroll
    for (int c = 0; c < KSPLIT; ++c)
        s += part[(size_t)c * PART_STRIDE + idx];   // fixed order -> deterministic
    out[idx] = s;
}

// ---------------- launcher ----------------
extern "C" void kernel_launch(void* const* d_in, const int* in_sizes, int n_in,
                              void* d_out, int out_size, void* d_ws, size_t ws_size,
                              hipStream_t stream) {
    const float* x    = (const float*)d_in[0];
    const float* wpos = (const float*)d_in[1];
    const float* wneg = (const float*)d_in[2];
    const float* bpos = (const float*)d_in[3];
    const float* bneg = (const float*)d_in[4];
    const float* npar = (const float*)d_in[5];
    float* out  = (float*)d_out;
    unsigned* wsu = (unsigned*)d_ws;
    float* af   = (float*)((char*)d_ws + AF_OFF);
    float* part = (float*)((char*)d_ws + PART_OFF);   // ~5.2 MB of ws total

    init_ws_k<<<1, 1, 0, stream>>>(wsu);
    maxabs_k<<<512, 256, 0, stream>>>(wpos, wneg, wsu);

    build_afrag_k<<<((int)AF_FLTS + 255) / 256, 256, 0, stream>>>(x, wsu, af);

    // 256 blocks x 4 waves = 1024 waves (64 stripes x 16 K-chunks)
    mem_gemm_k<<<16 * KSPLIT, 128, 0, stream>>>(wpos, wneg, bpos, bneg, npar,
                                                af, wsu, part);

    reduce_out_k<<<(PART_STRIDE + 255) / 256, 256, 0, stream>>>(part, out);
}